// RSNNCore_70609262346748
// MI455X (gfx1250) — compile-verified
//
#include <hip/hip_runtime.h>

typedef __attribute__((ext_vector_type(2))) float v2f;
typedef __attribute__((ext_vector_type(4))) float v4f;
typedef __attribute__((ext_vector_type(8))) float v8f;

#define B_  128
#define T_  500
#define I_  512
#define H_  1024
#define O_  11
#define BETA_   0.9f
#define THRESH_ 1.0f
#define P_  34   // transposed-LDS column pitch (dwords). Even -> every (col*P + k)*4
                 // is 8B-aligned so a B fragment is one ds_load_b64; 34 spreads banks.

// Workspace layout (floats):
//   Weff [H_*H_]  | v [B_*H_] | s0 [B_*H_] | s1 [B_*H_] | ssum [B_*H_]

__global__ __launch_bounds__(256)
void rsnn_init_kernel(const float* __restrict__ Wrec, const float* __restrict__ mask,
                      float* __restrict__ Weff, float* __restrict__ state) {
    int i = blockIdx.x * blockDim.x + threadIdx.x;
    if (i < H_ * H_)      Weff[i]  = Wrec[i] * mask[i];
    if (i < 4 * B_ * H_)  state[i] = 0.0f;
}

// One LIF timestep. Block = 32 H-columns (n0), 8 waves = 8 batch tiles of 16.
// Weight chunks (32K x 32N, Win rows then Weff rows) are staged TRANSPOSED into
// LDS (ldsT[col][k], double-buffered) and shared by all 8 waves. A lane's B
// fragment {B[k][n], B[k+1][n]} is then contiguous in LDS -> one ds_load_b64
// into an even-aligned VGPR pair (no repacking movs before v_wmma).
// f32 WMMA fragment layout (ISA 7.12.2):
//   A 16x4 : lane = half*16 + m ; VGPR j holds A[m][half*2 + j]   (m = lane&15)
//   B 4x16 : lane = half*16 + n ; VGPR j holds B[half*2 + j][n]   (n = lane&15)
//   C 16x16: lane = half*16 + n ; VGPR i holds C[half*8 + i][n]
__global__ __launch_bounds__(256)
void rsnn_step_kernel(const float* __restrict__ x,     const float* __restrict__ Win,
                      const float* __restrict__ Weff,  const float* __restrict__ hbias,
                      float* __restrict__ v,           const float* __restrict__ s_in,
                      float* __restrict__ s_out,       float* __restrict__ ssum,
                      int t) {
    __shared__ float lds[2][32 * P_];

    const int lane  = threadIdx.x & 31;
    const int wave  = threadIdx.x >> 5;        // batch tile 0..7
    const int n0    = blockIdx.x * 32;         // H columns owned by this block
    const int m0    = wave * 16;
    const int l     = lane & 15;
    const int half  = lane >> 4;
    const int kk    = half * 2;                // K sub-offset inside a 4-wide WMMA step
    const int mbase = half * 8;                // C/D row base

    const int rr = threadIdx.x >> 3;           // staging K-row 0..31
    const int cc = (threadIdx.x & 7) * 4;      // staging col 0,4,..,28

    const float* xrow = x    + ((size_t)(m0 + l) * T_ + t) * I_;
    const float* srow = s_in +  (size_t)(m0 + l) * H_;

    v8f acc0 = {};
    v8f acc1 = {};

    // Stage chunk 0 (Win rows 0..31), transposed: ldsT[col][k].
    {
        v4f w = *(const v4f*)(Win + (size_t)rr * H_ + n0 + cc);
        float* d = &lds[0][cc * P_ + rr];
        d[0] = w.x;  d[P_] = w.y;  d[2 * P_] = w.z;  d[3 * P_] = w.w;
    }
    __syncthreads();

    const int NCHI = I_ / 32;                  // 16 chunks of input-drive K
    const int NCH  = (I_ + H_) / 32;           // + 32 chunks of recurrent K = 48
    for (int c = 0; c < NCH; ++c) {
        const int cur = c & 1;
        const int nxt = cur ^ 1;

        // Prefetch next weight chunk into the other LDS buffer (uniform branch).
        if (c + 1 < NCH) {
            const int cn = c + 1;
            const float* W  = (cn < NCHI) ? Win : Weff;
            const int    kc = (cn < NCHI) ? cn * 32 : (cn - NCHI) * 32;
            v4f w = *(const v4f*)(W + (size_t)(kc + rr) * H_ + n0 + cc);
            float* d = &lds[nxt][cc * P_ + rr];
            d[0] = w.x;  d[P_] = w.y;  d[2 * P_] = w.z;  d[3 * P_] = w.w;
        }

        // Compute on the current chunk.
        const float* Arow = (c < NCHI) ? xrow : srow;
        const int    ka   = (c < NCHI) ? c * 32 : (c - NCHI) * 32;
        const float* buf  = lds[cur];
        #pragma unroll
        for (int k4 = 0; k4 < 32; k4 += 4) {
            v2f a  = *(const v2f*)(Arow + ka + k4 + kk);
            v2f b0 = *(const v2f*)(buf + (size_t)l        * P_ + k4 + kk);  // ds_load_b64
            v2f b1 = *(const v2f*)(buf + (size_t)(l + 16) * P_ + k4 + kk);  // ds_load_b64
            acc0 = __builtin_amdgcn_wmma_f32_16x16x4_f32(false, a, false, b0, (short)0, acc0, false, false);
            acc1 = __builtin_amdgcn_wmma_f32_16x16x4_f32(false, a, false, b1, (short)0, acc1, false, false);
        }
        __syncthreads();
    }

    // ---- fused LIF update + spike + reset + rate accumulation ----
    const float hb0 = hbias[n0 + l];
    const float hb1 = hbias[n0 + 16 + l];
    #pragma unroll
    for (int i = 0; i < 8; ++i) {
        const int bb = m0 + mbase + i;             // batch row owned by this lane/VGPR
        {
            const size_t idx = (size_t)bb * H_ + n0 + l;
            float vn = BETA_ * v[idx] + acc0[i] + hb0;
            float sp = (vn - THRESH_ > 0.0f) ? 1.0f : 0.0f;
            vn -= sp * THRESH_;
            v[idx] = vn;  s_out[idx] = sp;  ssum[idx] += sp;
        }
        {
            const size_t idx = (size_t)bb * H_ + n0 + 16 + l;
            float vn = BETA_ * v[idx] + acc1[i] + hb1;
            float sp = (vn - THRESH_ > 0.0f) ? 1.0f : 0.0f;
            vn -= sp * THRESH_;
            v[idx] = vn;  s_out[idx] = sp;  ssum[idx] += sp;
        }
    }
}

// out[b,o] = (ssum[b,:] / T) @ Wout[:,o] + bout[o]   (tiny: 128x11, K=1024)
__global__ __launch_bounds__(256)
void rsnn_readout_kernel(const float* __restrict__ ssum, const float* __restrict__ Wout,
                         const float* __restrict__ bout, float* __restrict__ out) {
    int idx = blockIdx.x * blockDim.x + threadIdx.x;
    if (idx >= B_ * O_) return;
    const int bb = idx / O_;
    const int o  = idx % O_;
    const float* sr = ssum + (size_t)bb * H_;
    float acc = 0.0f;
    for (int h = 0; h < H_; ++h)
        acc = fmaf(sr[h], Wout[h * O_ + o], acc);
    out[idx] = acc * (1.0f / (float)T_) + bout[o];
}

extern "C" void kernel_launch(void* const* d_in, const int* in_sizes, int n_in,
                              void* d_out, int out_size, void* d_ws, size_t ws_size,
                              hipStream_t stream) {
    (void)in_sizes; (void)n_in; (void)out_size; (void)ws_size;

    const float* x     = (const float*)d_in[0];   // [B,T,I]
    const float* Win   = (const float*)d_in[1];   // [I,H]
    const float* Wrec  = (const float*)d_in[2];   // [H,H]
    const float* Wout  = (const float*)d_in[3];   // [H,O]
    const float* hbias = (const float*)d_in[4];   // [H]
    const float* bout  = (const float*)d_in[5];   // [O]
    const float* mask  = (const float*)d_in[6];   // [H,H]
    float* out = (float*)d_out;                   // [B,O]

    float* ws   = (float*)d_ws;
    float* Weff = ws;                             // H*H
    float* v    = Weff + (size_t)H_ * H_;         // B*H
    float* s0   = v    + (size_t)B_ * H_;         // B*H
    float* s1   = s0   + (size_t)B_ * H_;         // B*H
    float* ssum = s1   + (size_t)B_ * H_;         // B*H

    // Recomputed every launch: deterministic, no cross-call state.
    rsnn_init_kernel<<<(H_ * H_ + 255) / 256, 256, 0, stream>>>(Wrec, mask, Weff, v);

    for (int t = 0; t < T_; ++t) {
        const float* sin  = (t & 1) ? s1 : s0;    // double-buffered spikes: no RW race
        float*       sout = (t & 1) ? s0 : s1;
        rsnn_step_kernel<<<32, 256, 0, stream>>>(x, Win, Weff, hbias, v, sin, sout, ssum, t);
    }

    rsnn_readout_kernel<<<(B_ * O_ + 255) / 256, 256, 0, stream>>>(ssum, Wout, bout, out);
}